// LightGCN_9380208574850
// MI455X (gfx1250) — compile-verified
//
#include <hip/hip_runtime.h>

namespace {

constexpr int kNodes  = 50000;
constexpr int kDim    = 64;       // floats per node row = 256 B
constexpr int kEdges  = 1250000;

// ---------------------------------------------------------------- utilities
__global__ void zero_i32(int* __restrict__ p, int n) {
  int i = blockIdx.x * blockDim.x + threadIdx.x;
  if (i < n) p[i] = 0;
}

// ------------------------------------------------------- degree histogram
__global__ void count_deg(const int* __restrict__ row, int* __restrict__ deg, int e) {
  int i = blockIdx.x * blockDim.x + threadIdx.x;
  if (i < e) atomicAdd(&deg[row[i]], 1);
}

// ------------------- exclusive scan of deg -> rowptr, plus dinv = rsqrt(deg)
__global__ __launch_bounds__(1024)
void scan_rowptr_dinv(const int* __restrict__ deg, int* __restrict__ rowptr,
                      float* __restrict__ dinv, int n) {
  __shared__ int sums[1024];
  const int t     = threadIdx.x;
  const int chunk = (n + 1023) / 1024;
  const int start = t * chunk;
  const int end   = min(start + chunk, n);

  int s = 0;
  for (int i = start; i < end; ++i) s += deg[i];
  sums[t] = s;
  __syncthreads();

  for (int off = 1; off < 1024; off <<= 1) {   // Hillis-Steele inclusive scan
    int v = (t >= off) ? sums[t - off] : 0;
    __syncthreads();
    sums[t] += v;
    __syncthreads();
  }

  int run = (t == 0) ? 0 : sums[t - 1];
  for (int i = start; i < end; ++i) {
    rowptr[i] = run;
    const int d = deg[i];
    dinv[i] = (d > 0) ? rsqrtf((float)d) : 0.0f;
    run += d;
  }
  if (t == 1023) rowptr[n] = sums[1023];
}

// ----------------------------------------------------------- CSR fill
__global__ void fill_csr(const int* __restrict__ row, const int* __restrict__ col,
                         const int* __restrict__ rowptr, int* __restrict__ cnt,
                         int* __restrict__ cidx, int e) {
  int i = blockIdx.x * blockDim.x + threadIdx.x;
  if (i < e) {
    const int r = row[i];
    const int p = atomicAdd(&cnt[r], 1);
    cidx[rowptr[r] + p] = col[i];
  }
}

// Sort each row's neighbor list -> bitwise-deterministic summation order
// regardless of the atomic fill order. Avg degree 25; runs once per launch.
__global__ void sort_rows(const int* __restrict__ rowptr, int* __restrict__ cidx, int n) {
  int i = blockIdx.x * blockDim.x + threadIdx.x;
  if (i >= n) return;
  const int s = rowptr[i], e = rowptr[i + 1];
  for (int a = s + 1; a < e; ++a) {
    const int v = cidx[a];
    int b = a - 1;
    while (b >= s && cidx[b] > v) { cidx[b + 1] = cidx[b]; --b; }
    cidx[b + 1] = v;
  }
}

// ------------------------------------------------- per-layer aggregation
// One wave32 per node; lane owns a float2 slice of the 64-dim row so each
// neighbor gather is one full 256 B row per wave instruction.
//
// Per 32-neighbor chunk:
//   - lane loads 1 index           (coalesced: 1 VMEM load / 32 neighbors)
//   - lane prefetches ITS index's row (2x global_prefetch_b8 covers the
//     whole chunk's 32 gather rows before consumption starts)
//   - inner loop broadcasts lane j's index via __shfl (wave32 permute)
//
// FUSE_INIT=true (layer 0): acc = 0.25*emb + 0.25*x_new  (no acc read,
// replaces the separate init pass). Otherwise: acc += 0.25*x_new.
template <bool FUSE_INIT>
__global__ __launch_bounds__(256)
void aggregate(const float* __restrict__ xin, float* __restrict__ xout,
               float* __restrict__ acc, const int* __restrict__ rowptr,
               const int* __restrict__ cidx, const float* __restrict__ dinv,
               int n) {
  const int gid  = blockIdx.x * blockDim.x + threadIdx.x;
  const int node = gid >> 5;
  const int lane = gid & 31;
  if (node >= n) return;

  const int s = rowptr[node];
  const int e = rowptr[node + 1];

  float sx = 0.0f, sy = 0.0f;
  for (int k0 = s; k0 < e; k0 += 32) {
    const int kk = k0 + lane;
    int idx = 0;
    if (kk < e) {
      idx = cidx[kk];                                   // 32 indices, 1 load
      const float* rowp = &xin[(size_t)idx * kDim];
      __builtin_prefetch(rowp,      0, 1);              // global_prefetch_b8
      __builtin_prefetch(rowp + 32, 0, 1);              // second 128 B half
    }
    const int m = min(32, e - k0);
#pragma unroll 4
    for (int j = 0; j < m; ++j) {
      const int c = __shfl(idx, j, 32);                 // wave32 broadcast
      const float2 v =
          *reinterpret_cast<const float2*>(&xin[(size_t)c * kDim + lane * 2]);
      sx += v.x;
      sy += v.y;
    }
  }

  const float d  = dinv[node];
  const float sc = d * d;                   // dinv[row] on msg * dinv on result
  const float2 o = make_float2(sx * sc, sy * sc);

  const size_t base = (size_t)node * kDim + lane * 2;
  *reinterpret_cast<float2*>(&xout[base]) = o;

  float2 a;
  if (FUSE_INIT) {
    // layer 0: acc starts at 0.25*emb; xin IS emb here, same row layout.
    const float2 e0 = *reinterpret_cast<const float2*>(&xin[base]);
    a.x = 0.25f * e0.x + 0.25f * o.x;       // *0.25 exact == final /4
    a.y = 0.25f * e0.y + 0.25f * o.y;
  } else {
    a = *reinterpret_cast<const float2*>(&acc[base]);
    a.x += 0.25f * o.x;
    a.y += 0.25f * o.y;
  }
  *reinterpret_cast<float2*>(&acc[base]) = a;
}

inline size_t align256(size_t x) { return (x + 255) & ~size_t(255); }

} // namespace

extern "C" void kernel_launch(void* const* d_in, const int* in_sizes, int n_in,
                              void* d_out, int out_size, void* d_ws, size_t ws_size,
                              hipStream_t stream) {
  (void)in_sizes; (void)n_in; (void)out_size; (void)ws_size;

  const int*   edge = (const int*)d_in[0];      // [2, E] int32
  const float* emb  = (const float*)d_in[1];    // [N, D] float32
  const int*   row  = edge;
  const int*   col  = edge + kEdges;
  float*       acc  = (float*)d_out;            // [N, D] float32 output / accumulator

  // ---- workspace layout (all 256B aligned) : ~31.4 MB total
  char* ws = (char*)d_ws;
  size_t off = 0;
  int*   deg    = (int*)(ws + off);   off += align256(sizeof(int)   * kNodes);
  int*   cnt    = (int*)(ws + off);   off += align256(sizeof(int)   * kNodes);
  int*   rowptr = (int*)(ws + off);   off += align256(sizeof(int)   * (kNodes + 1));
  float* dinv   = (float*)(ws + off); off += align256(sizeof(float) * kNodes);
  int*   cidx   = (int*)(ws + off);   off += align256(sizeof(int)   * kEdges);
  float* xA     = (float*)(ws + off); off += align256(sizeof(float) * (size_t)kNodes * kDim);
  float* xB     = (float*)(ws + off); off += align256(sizeof(float) * (size_t)kNodes * kDim);

  // ---- build normalization + CSR (deterministic end state)
  zero_i32<<<(kNodes + 255) / 256, 256, 0, stream>>>(deg, kNodes);
  zero_i32<<<(kNodes + 255) / 256, 256, 0, stream>>>(cnt, kNodes);
  count_deg<<<(kEdges + 255) / 256, 256, 0, stream>>>(row, deg, kEdges);
  scan_rowptr_dinv<<<1, 1024, 0, stream>>>(deg, rowptr, dinv, kNodes);
  fill_csr<<<(kEdges + 255) / 256, 256, 0, stream>>>(row, col, rowptr, cnt, cidx, kEdges);
  sort_rows<<<(kNodes + 255) / 256, 256, 0, stream>>>(rowptr, cidx, kNodes);

  // ---- 3 propagation layers (ping-pong x buffers), one wave per node.
  // Layer 0 fuses acc = 0.25*emb (layer-0 term) with its own 0.25*x1 term.
  const int aggBlocks = (kNodes + 7) / 8;   // 8 waves (nodes) per 256-thread block
  aggregate<true ><<<aggBlocks, 256, 0, stream>>>(emb, xA, acc, rowptr, cidx, dinv, kNodes);
  aggregate<false><<<aggBlocks, 256, 0, stream>>>(xA,  xB, acc, rowptr, cidx, dinv, kNodes);
  aggregate<false><<<aggBlocks, 256, 0, stream>>>(xB,  xA, acc, rowptr, cidx, dinv, kNodes);
}